// EdgeNet_26053271618005
// MI455X (gfx1250) — compile-verified
//
#include <hip/hip_runtime.h>
#include <hip/hip_bf16.h>

// ---------------------------------------------------------------------------
// Problem constants (from the reference)
// ---------------------------------------------------------------------------
#define NQ        16
#define NSTATE    65536     // 2^16 amplitudes
#define NEDGE     128       // E
#define NSAMP     512       // N
#define NFEAT     8         // F
#define MEAS_BIT  5         // wire 10 -> bit position (15 - 10) from LSB

// Gate list: type 0 = ry(wire=a, theta_idx=b), type 1 = cx(control=a, target=b)
#define NGATES 46
__device__ __constant__ const unsigned char G_TYPE[NGATES] = {
  0,0,1, 0,0,1, 0,0,1, 0,0,1, 0,0,1, 0,0,1, 0,0,1, 0,0,1,
  0,0,1, 0,0,1, 0,0,1, 0,0,1,
  0,0,1, 0,0,1,
  0,0,1, 0};
__device__ __constant__ const unsigned char G_A[NGATES] = {
  0,1,0,  2,3,3,  4,5,4,  6,7,7,  8,9,8,  10,11,11, 12,13,12, 14,15,15,
  1,2,1,  5,6,6,  9,10,9, 13,14,14,
  2,5,2,  10,13,13,
  5,10,5, 10};
__device__ __constant__ const unsigned char G_B[NGATES] = {
  0,1,1,  2,3,2,  4,5,5,  6,7,6,  8,9,9,  10,11,10, 12,13,13, 14,15,14,
  16,17,2, 18,19,5, 20,21,10, 22,23,13,
  24,25,5, 26,27,10,
  28,29,10, 30};

typedef __attribute__((ext_vector_type(2))) float        v2f;
typedef __attribute__((ext_vector_type(8))) float        v8f;
typedef __attribute__((ext_vector_type(4))) unsigned int u32x4;
typedef __attribute__((ext_vector_type(8))) unsigned int u32x8;

// ---------------------------------------------------------------------------
// Kernel 1: B = [Ro^T X | Ri^T X]  (128 x 16) via V_WMMA_F32_16X16X4_F32.
// One block of 512 threads = 16 wave32 waves:
//   waves 0..7  : bo tiles (rows 16w .. 16w+15), cols 0..7   (Ro)
//   waves 8..15 : bi tiles,                      cols 8..15  (Ri)
// X (512x8 f32, 16 KB) is DMA'd into a zero-padded LDS tile Xpad[512][16]
// by the Tensor Data Mover (pad_interval=8 DWORDs, pad_amount=8 DWORDs ->
// row stride 16 floats), so the WMMA inner loop is branch-free.
// ---------------------------------------------------------------------------
__global__ __launch_bounds__(512) void edgenet_gemmB_wmma(
    const float* __restrict__ X,    // [NSAMP][NFEAT]
    const float* __restrict__ Ri,   // [NSAMP][NEDGE]
    const float* __restrict__ Ro,   // [NSAMP][NEDGE]
    float* __restrict__ Bmat)       // [NEDGE][16]
{
  __shared__ float Xpad[NSAMP][16];   // 32 KB: cols 0..7 data, 8..15 zeros

  const int tid  = threadIdx.x;
  const int wave = tid >> 5;          // 0..15
  const int lane = tid & 31;
  const int half = lane >> 4;         // 0: lanes 0-15, 1: lanes 16-31
  const int l    = lane & 15;

  // Pre-zero only the padding columns (no overlap with the TDM destination).
  for (int i = tid; i < NSAMP * NFEAT; i += 512) {
    Xpad[i >> 3][8 + (i & 7)] = 0.0f;
  }
  __syncthreads();

  // --- TDM: tensor_load_to_lds of 4096 dwords with LDS padding ------------
  if (wave == 0) {
    const unsigned long long ga = (unsigned long long)(uintptr_t)X;
    const unsigned int lds0 = (unsigned int)(uintptr_t)&Xpad[0][0];
    u32x4 g0;
    g0[0] = 1u;                                   // count=1, user load desc
    g0[1] = lds0;                                 // lds_addr
    g0[2] = (unsigned int)ga;                     // global_addr[31:0]
    g0[3] = ((unsigned int)(ga >> 32) & 0x01FFFFFFu) | 0x80000000u; // [56:32] | type=2
    u32x8 g1;
    g1[0] = (2u << 16)      // data_size = 4 bytes
          | (1u << 20)      // pad_enable
          | (2u << 22)      // pad_interval: every 8 DWORDs
          | (7u << 25);     // pad_amount: 8 DWORDs
    g1[1] = (4096u & 0xFFFFu) << 16;  // tensor_dim0 = 4096 (bits 79:48)
    g1[2] = 1u << 16;                 // tensor_dim1 = 1    (bits 111:80)
    g1[3] = 4096u << 16;              // tile_dim0  = 4096  (bits 127:112)
    g1[4] = 1u;                       // tile_dim1  = 1
    g1[5] = 4096u;                    // tensor_dim0_stride = 4096
    g1[6] = 0u;
    g1[7] = 0u;
    asm volatile("tensor_load_to_lds %0, %1" : : "s"(g0), "s"(g1) : "memory");
    __builtin_amdgcn_s_wait_tensorcnt(0);
  }
  __syncthreads();

  const float* __restrict__ R = (wave < 8) ? Ro : Ri;
  const int mt     = wave & 7;        // edge tile
  const int e0     = mt * 16;
  const int colOff = (wave < 8) ? 0 : 8;

  v8f acc = {};
  // A[m][k] = R[k][e0+m]   (ISA 16x4 f32 A layout: half selects the K pair)
  // B[k][n] = Xpad[k][n]   (zero-padded, branch-free ds_load)
  for (int k0 = 0; k0 < NSAMP; k0 += 4) {
    const int ka = k0 + half * 2;
    v2f a, b;
    a.x = R[(ka + 0) * NEDGE + e0 + l];
    a.y = R[(ka + 1) * NEDGE + e0 + l];
    b.x = Xpad[ka + 0][l];
    b.y = Xpad[ka + 1][l];
    acc = __builtin_amdgcn_wmma_f32_16x16x4_f32(
        /*neg_a=*/false, a, /*neg_b=*/false, b,
        /*c_mod=*/(short)0, acc, /*reuse_a=*/false, /*reuse_b=*/false);
  }

  // C/D layout: VGPR j -> M = j + 8*half, N = l  (ISA 16x16 f32 C layout)
  if (l < NFEAT) {
#pragma unroll
    for (int j = 0; j < 8; ++j) {
      const int m = half * 8 + j;
      Bmat[(e0 + m) * 16 + colOff + l] = acc[j];
    }
  }
}

// ---------------------------------------------------------------------------
// Kernel 2: statevector circuit, one workgroup per edge.
// State (65536 f32 = 256 KB) lives entirely in LDS (MI455X WGP has 320 KB).
// 1024 threads = 32 wave32 waves.
// ---------------------------------------------------------------------------
__global__ __launch_bounds__(1024) void edgenet_circuit(
    const float* __restrict__ Bmat,    // [NEDGE][16]
    const float* __restrict__ theta,   // [31]
    float* __restrict__ out)           // [NEDGE]
{
  extern __shared__ float state[];     // NSTATE floats (256 KB dynamic LDS)
  __shared__ float cw[NQ], sw[NQ];     // per-edge input rotations
  __shared__ float gc[31], gs[31];     // learned gate rotations
  __shared__ float red[32];

  const int e   = blockIdx.x;
  const int tid = threadIdx.x;

  if (tid < NQ) {
    const float h = 0.5f * Bmat[e * 16 + tid];
    cw[tid] = __cosf(h);
    sw[tid] = __sinf(h);
  }
  if (tid >= 32 && tid < 32 + 31) {
    const int i = tid - 32;
    const float h = 0.5f * theta[i];
    gc[i] = __cosf(h);
    gs[i] = __sinf(h);
  }
  __syncthreads();

  // --- initial product state: amp(i) = prod_w (bit ? sin : cos) -----------
  for (int idx = tid; idx < NSTATE; idx += 1024) {
    float amp = 1.0f;
#pragma unroll
    for (int w = 0; w < NQ; ++w)
      amp *= ((idx >> (15 - w)) & 1) ? sw[w] : cw[w];
    state[idx] = amp;
  }
  __syncthreads();

  // --- gates ----------------------------------------------------------------
  for (int g = 0; g < NGATES; ++g) {
    if (G_TYPE[g] == 0) {
      // RY on wire a with angle theta[b]
      const int k      = 15 - (int)G_A[g];
      const int stride = 1 << k;
      const float c = gc[G_B[g]];
      const float s = gs[G_B[g]];
      for (int p = tid; p < (NSTATE >> 1); p += 1024) {
        const int i0 = ((p >> k) << (k + 1)) | (p & (stride - 1));
        const int i1 = i0 | stride;
        const float a0 = state[i0];
        const float a1 = state[i1];
        state[i0] = c * a0 - s * a1;
        state[i1] = s * a0 + c * a1;
      }
    } else {
      // CNOT control=a, target=b: swap target bit where control bit == 1
      const int kc = 15 - (int)G_A[g];
      const int kt = 15 - (int)G_B[g];
      const int k1 = (kc < kt) ? kc : kt;
      const int k2 = (kc < kt) ? kt : kc;
      const int cmask = 1 << kc;
      const int tmask = 1 << kt;
      for (int p = tid; p < (NSTATE >> 2); p += 1024) {
        const int low  = p & ((1 << k1) - 1);
        const int mid  = (p >> k1) & ((1 << (k2 - 1 - k1)) - 1);
        const int high = p >> (k2 - 1);
        const int i = (high << (k2 + 1)) | (mid << (k1 + 1)) | low | cmask;
        const float t0 = state[i];
        const float t1 = state[i | tmask];
        state[i]         = t1;
        state[i | tmask] = t0;
      }
    }
    __syncthreads();
  }

  // --- measurement: z = P(bit5==0) - P(bit5==1); out = (1 - z)/2 ----------
  float d = 0.0f;
  for (int idx = tid; idx < NSTATE; idx += 1024) {
    float v = state[idx];
    v *= v;
    d += (idx & (1 << MEAS_BIT)) ? -v : v;
  }
#pragma unroll
  for (int off = 16; off > 0; off >>= 1)
    d += __shfl_down(d, off, 32);            // wave32 reduction
  if ((tid & 31) == 0) red[tid >> 5] = d;
  __syncthreads();
  if (tid == 0) {
    float z = 0.0f;
#pragma unroll
    for (int i = 0; i < 32; ++i) z += red[i];
    out[e] = 0.5f * (1.0f - z);
  }
}

// ---------------------------------------------------------------------------
// Host entry
// ---------------------------------------------------------------------------
extern "C" void kernel_launch(void* const* d_in, const int* in_sizes, int n_in,
                              void* d_out, int out_size, void* d_ws, size_t ws_size,
                              hipStream_t stream) {
  const float* X     = (const float*)d_in[0];   // [512][8]
  const float* Ri    = (const float*)d_in[1];   // [512][128]
  const float* Ro    = (const float*)d_in[2];   // [512][128]
  const float* theta = (const float*)d_in[3];   // [31]
  float* out  = (float*)d_out;                  // [128]
  float* Bmat = (float*)d_ws;                   // [128][16] scratch (8 KB)

  edgenet_gemmB_wmma<<<1, 512, 0, stream>>>(X, Ri, Ro, Bmat);
  edgenet_circuit<<<NEDGE, 1024, NSTATE * sizeof(float), stream>>>(Bmat, theta, out);
}